// Scattering_57234734187254
// MI455X (gfx1250) — compile-verified
//
#include <hip/hip_runtime.h>
#include <hip/hip_bf16.h>

typedef __attribute__((ext_vector_type(2))) float v2f;
typedef __attribute__((ext_vector_type(8))) float v8f;

#define T_LEN   65536
#define TO_LEN  32769     // conv output length (stride 2, sym pad 55/57)
#define TP_LEN  8192      // pooled length
#define NFT     112       // filter taps
#define NCH     64        // 2*NSC channels
#define BC_N    48        // B*C

// d_out layout (floats): out_u | out_s | loss
#define OUT_S_OFF 50333184   // 16*32*3*32769
#define LOSS_IDX  62916096   // + 48*32*8192

// ws layout (floats):
//  [0, 7168)      kplain[w][o]           (112 x 64)
//  [7168, 14336)  kpair  float2[(w/2)*64+o] = (k[w][o], k[w+1][o])
//  [14336, 14782) H[2][223] inverse FIR taps
#define WS_KPAIR 7168
#define WS_H     14336

__device__ __forceinline__ int reflect_idx(int ip) {
  if (ip < 0) ip = -1 - ip;
  if (ip >= T_LEN) ip = 2 * T_LEN - 1 - ip;
  return ip;
}

// ---------------------------------------------------------------------------
// Kernel 1: build the 112x64 filter bank (Hermite interpolation) + zero loss.
// One block, 64 threads; thread o = i*32 + s.
// ---------------------------------------------------------------------------
__global__ void filters_kernel(const float* __restrict__ m, const float* __restrict__ p,
                               const float* __restrict__ sd, const float* __restrict__ kb,
                               float* __restrict__ ws, float* __restrict__ out) {
  const int o = threadIdx.x;
  if (o == 0) out[LOSS_IDX] = 0.f;
  if (o >= NCH) return;
  const int i = o >> 5, s = o & 31;

  const float scale = exp2f((float)s * 0.125f) + sd[s];
  float knots[7];
  float run = 0.f;
  for (int k = 0; k < 7; ++k) {
    float c = kb[k] * scale;
    c = fminf(fmaxf(c, 1.f), 105.f);            // clip to [1, NF-K]
    knots[k] = run - 3.f * scale;               // exclusive cumsum - (K//2)*scale
    run += c;
  }
  float mmean = 0.f;
  for (int k = 1; k <= 5; ++k) mmean += m[i * 7 + k];
  mmean *= 0.2f;
  float mv[7], pv[7];
  for (int k = 0; k < 7; ++k) {
    const float msk = (k == 0 || k == 6) ? 0.f : 1.f;
    mv[k] = (m[i * 7 + k] - mmean) * msk;
    pv[k] = p[i * 7 + k] * msk;
  }
  // cubic coeffs per interval from reference's (odd) Hermite pairing:
  // y = [m_t, m_{t+1}, p_t, p_{t+1}],  h rows [1,0,-3,2][0,1,-2,1][0,0,3,-2][0,0,-1,1]
  float c0[6], c1[6], c2[6], c3[6];
  for (int t = 0; t < 6; ++t) {
    const float y0 = mv[t], y1 = mv[t + 1], y2 = pv[t], y3 = pv[t + 1];
    c0[t] = y0;
    c1[t] = y1;
    c2[t] = -3.f * y0 - 2.f * y1 + 3.f * y2 - y3;
    c3[t] =  2.f * y0 +       y1 - 2.f * y2 + y3;
  }
  float vals[NFT];
  float maxv = -3.0e38f;
  for (int f = 0; f < NFT; ++f) {
    const float xi = -56.f + (float)f * (112.f / 111.f);  // linspace(-56,56,112)
    float v = 0.f;
    for (int t = 0; t < 6; ++t) {
      const float xn = (xi - knots[t]) / (knots[t + 1] - knots[t]);
      if (xn >= 0.f && xn < 1.f)
        v += c0[t] + xn * (c1[t] + xn * (c2[t] + xn * c3[t]));
    }
    vals[f] = v;
    maxv = fmaxf(maxv, v);
  }
  const float inv = 1.f / maxv;
  for (int f = 0; f < NFT; ++f) ws[f * NCH + o] = vals[f] * inv;
  float2* kp = (float2*)(ws + WS_KPAIR);
  for (int kk = 0; kk < NFT / 2; ++kk) {
    float2 v; v.x = vals[2 * kk] * inv; v.y = vals[2 * kk + 1] * inv;
    kp[kk * NCH + o] = v;
  }
}

// ---------------------------------------------------------------------------
// Kernel 2: interior inverse FIR taps  H_par[delta] = sum_{d==par} k[d,:].k[d-delta,:]
// ---------------------------------------------------------------------------
__global__ void h_kernel(float* __restrict__ ws) {
  for (int idx = threadIdx.x; idx < 2 * 223; idx += 256) {
    const int par = idx / 223;
    const int dlt = idx % 223 - 111;
    float s = 0.f;
    for (int d = par; d < NFT; d += 2) {
      const int w2 = d - dlt;
      if (w2 < 0 || w2 >= NFT) continue;
      const float* ka = ws + d * NCH;
      const float* kb = ws + w2 * NCH;
      for (int o = 0; o < NCH; ++o) s = fmaf(ka[o], kb[o], s);
    }
    ws[WS_H + idx] = s;
  }
}

// ---------------------------------------------------------------------------
// Kernel 3: strided conv as WMMA f32 16x16x4 GEMM + modulus + out_u + pooled out_s.
// Block = 8 waves, each wave: 16 output positions x 64 channels, K=112 in 28 steps.
// ---------------------------------------------------------------------------
__global__ __launch_bounds__(256) void conv_kernel(const float* __restrict__ x,
                                                   const float* __restrict__ ws,
                                                   float* __restrict__ out) {
  __shared__ v2f  klds[56 * NCH];        // (k,k+1) channel pairs, 28 KB
  __shared__ float xpl[2 * 128 + NFT + 8];
  const int bc = blockIdx.y;
  const int to_blk = blockIdx.x * 128;
  const int tid = threadIdx.x;
  const int wave = tid >> 5, lane = tid & 31;
  const int row = lane & 15, half = lane >> 4;

  const v2f* kp = (const v2f*)(ws + WS_KPAIR);
  for (int i = tid; i < 56 * NCH; i += 256) klds[i] = kp[i];
  const float* xb = x + (size_t)bc * T_LEN;
  for (int j = tid; j < 2 * 128 + NFT; j += 256) {
    int ip = reflect_idx(2 * to_blk + j - 55);
    ip = (ip < 0) ? 0 : (ip >= T_LEN ? T_LEN - 1 : ip);   // clamp for overrun tiles
    xpl[j] = xb[ip];
  }
  __syncthreads();

  v8f acc0 = {0.f,0.f,0.f,0.f,0.f,0.f,0.f,0.f};
  v8f acc1 = acc0, acc2 = acc0, acc3 = acc0;
  const int abase = 2 * (wave * 16 + row) + 2 * half;   // A[m][k]: xp[2*(to0+m)+k]
  for (int ks = 0; ks < 28; ++ks) {
    const int k0 = ks * 4;
    const v2f a = *(const v2f*)&xpl[abase + k0];
    const int kb0 = ((k0 >> 1) + half) * NCH + row;
    const v2f b0 = klds[kb0];
    const v2f b1 = klds[kb0 + 16];
    const v2f b2 = klds[kb0 + 32];
    const v2f b3 = klds[kb0 + 48];
    acc0 = __builtin_amdgcn_wmma_f32_16x16x4_f32(false, a, false, b0, (short)0, acc0, false, false);
    acc1 = __builtin_amdgcn_wmma_f32_16x16x4_f32(false, a, false, b1, (short)0, acc1, false, false);
    acc2 = __builtin_amdgcn_wmma_f32_16x16x4_f32(false, a, false, b2, (short)0, acc2, false, false);
    acc3 = __builtin_amdgcn_wmma_f32_16x16x4_f32(false, a, false, b3, (short)0, acc3, false, false);
  }

  // modulus: u[s] = sqrt(xc[s]^2 + xc[s+32]^2); lane holds col s=row (and s=row+16)
  float u0[8], u1[8];
#pragma unroll
  for (int r = 0; r < 8; ++r) {
    u0[r] = sqrtf(acc0[r] * acc0[r] + acc2[r] * acc2[r]);
    u1[r] = sqrtf(acc1[r] * acc1[r] + acc3[r] * acc3[r]);
  }

  const int to0 = to_blk + wave * 16;
  const int bq = bc / 3, cq = bc - 3 * bq;
  // out_u with the reference's flat (To,NSC)->(NSC,To) reinterpretation
#pragma unroll
  for (int r = 0; r < 8; ++r) {
    const int t = to0 + half * 8 + r;                 // D rows: lanes<16 -> 0..7, >=16 -> 8..15
    if (t < TO_LEN) {
      const int g0 = t * 32 + row;
      const int A0 = g0 / TO_LEN, d0 = g0 - A0 * TO_LEN;
      out[((bq * 32 + A0) * 3 + cq) * TO_LEN + d0] = u0[r];
      const int g1 = g0 + 16;
      const int A1 = g1 / TO_LEN, d1 = g1 - A1 * TO_LEN;
      out[((bq * 32 + A1) * 3 + cq) * TO_LEN + d1] = u1[r];
    }
  }
  // pooled: window 4 / stride 4 lives entirely inside one lane's regs
  const int tstart = to0 + half * 8;
  const int tp0 = tstart >> 2;
  if (tp0 < TP_LEN) {
    out[OUT_S_OFF + (bc * 32 + row) * TP_LEN + tp0]      = (u0[0] + u0[1] + u0[2] + u0[3]) * 0.25f;
    out[OUT_S_OFF + (bc * 32 + row + 16) * TP_LEN + tp0] = (u1[0] + u1[1] + u1[2] + u1[3]) * 0.25f;
  }
  if (tp0 + 1 < TP_LEN) {
    out[OUT_S_OFF + (bc * 32 + row) * TP_LEN + tp0 + 1]      = (u0[4] + u0[5] + u0[6] + u0[7]) * 0.25f;
    out[OUT_S_OFF + (bc * 32 + row + 16) * TP_LEN + tp0 + 1] = (u1[4] + u1[5] + u1[6] + u1[7]) * 0.25f;
  }
}

// ---------------------------------------------------------------------------
// Kernel 4: interior inverse = 223-tap FIR on reflected x; loss partial sums.
// Covers t in [64, T-64).
// ---------------------------------------------------------------------------
__global__ __launch_bounds__(256) void inv_interior_kernel(const float* __restrict__ x,
                                                           const float* __restrict__ ws,
                                                           float* __restrict__ out) {
  __shared__ float xs[480];
  __shared__ float Hl[446];
  __shared__ float red[256];
  const int bc = blockIdx.y;
  const int t0 = 64 + blockIdx.x * 256;
  const float* xb = x + (size_t)bc * T_LEN;
  for (int i = threadIdx.x; i < 446; i += 256) Hl[i] = ws[WS_H + i];
  for (int j = threadIdx.x; j < 478; j += 256) xs[j] = xb[reflect_idx(t0 - 111 + j)];
  __syncthreads();
  const int t = t0 + threadIdx.x;
  float sq = 0.f;
  if (t < T_LEN - 64) {
    const float* Hp = &Hl[((t + 55) & 1) * 223];
    const float* xv = &xs[threadIdx.x];
    float g = 0.f;
    for (int j = 0; j < 223; ++j) g = fmaf(Hp[j], xv[222 - j], g);
    const float diff = g - xs[threadIdx.x + 111];
    sq = diff * diff;
  }
  red[threadIdx.x] = sq;
  __syncthreads();
  for (int off = 128; off > 0; off >>= 1) {
    if (threadIdx.x < off) red[threadIdx.x] += red[threadIdx.x + off];
    __syncthreads();
  }
  if (threadIdx.x == 0) atomicAdd(out + LOSS_IDX, red[0] * (1.f / 3145728.f));
}

// ---------------------------------------------------------------------------
// Kernel 5: exact boundary inverse for t in [0,64) u [T-64,T): recompute the
// needed x_conv rows in LDS, apply the clipped transposed conv + pad fold.
// One block per bc (48 blocks, 128 threads).
// ---------------------------------------------------------------------------
__global__ __launch_bounds__(128) void inv_boundary_kernel(const float* __restrict__ x,
                                                           const float* __restrict__ ws,
                                                           float* __restrict__ out) {
  __shared__ float kl[NFT * NCH];      // 28 KB
  __shared__ float xcL[60 * NCH];      // x_conv rows to = 0..59
  __shared__ float xcR[61 * NCH];      // x_conv rows to = 32708..32768
  __shared__ float red[128];
  const int bc = blockIdx.x;
  const int tid = threadIdx.x;
  const float* xb = x + (size_t)bc * T_LEN;
  for (int i = tid; i < NFT * NCH; i += 128) kl[i] = ws[i];
  __syncthreads();
  for (int idx = tid; idx < (60 + 61) * NCH; idx += 128) {
    const int rr = idx >> 6, o = idx & 63;
    const int to = (rr < 60) ? rr : (32708 + (rr - 60));
    float s = 0.f;
    for (int w = 0; w < NFT; ++w)
      s = fmaf(xb[reflect_idx(2 * to + w - 55)], kl[w * NCH + o], s);
    if (rr < 60) xcL[rr * NCH + o] = s; else xcR[(rr - 60) * NCH + o] = s;
  }
  __syncthreads();

  auto gxp = [&](int n) -> float {
    int dmin = n - 2 * (TO_LEN - 1);
    if (dmin < 0) dmin = 0;
    int d = dmin + ((dmin ^ n) & 1);            // d == n (mod 2), d >= dmin
    float s = 0.f;
    for (; d < NFT && d <= n; d += 2) {
      const int to = (n - d) >> 1;
      const float* xc = (to < 60) ? &xcL[to * NCH] : &xcR[(to - 32708) * NCH];
      const float* kr = &kl[d * NCH];
      for (int o = 0; o < NCH; ++o) s = fmaf(xc[o], kr[o], s);
    }
    return s;
  };

  const int t = (tid < 64) ? tid : (T_LEN - 64 + (tid - 64));
  float invv = gxp(t + 55);
  if (t <= 54) invv += gxp(54 - t);                              // left sym-pad fold
  if (t >= T_LEN - 57) invv += gxp(55 + T_LEN + (T_LEN - 1 - t)); // right sym-pad fold
  const float diff = invv - xb[t];
  red[tid] = diff * diff;
  __syncthreads();
  for (int off = 64; off > 0; off >>= 1) {
    if (tid < off) red[tid] += red[tid + off];
    __syncthreads();
  }
  if (tid == 0) atomicAdd(out + LOSS_IDX, red[0] * (1.f / 3145728.f));
}

// ---------------------------------------------------------------------------
extern "C" void kernel_launch(void* const* d_in, const int* in_sizes, int n_in,
                              void* d_out, int out_size, void* d_ws, size_t ws_size,
                              hipStream_t stream) {
  (void)in_sizes; (void)n_in; (void)out_size; (void)ws_size;
  const float* x  = (const float*)d_in[0];
  const float* m  = (const float*)d_in[1];
  const float* p  = (const float*)d_in[2];
  const float* sd = (const float*)d_in[3];
  const float* kb = (const float*)d_in[4];
  float* out = (float*)d_out;
  float* ws  = (float*)d_ws;

  filters_kernel<<<1, 64, 0, stream>>>(m, p, sd, kb, ws, out);
  h_kernel<<<1, 256, 0, stream>>>(ws);
  conv_kernel<<<dim3(257, BC_N), 256, 0, stream>>>(x, ws, out);       // 257*8 waves >= 2049 tiles
  inv_interior_kernel<<<dim3(256, BC_N), 256, 0, stream>>>(x, ws, out);
  inv_boundary_kernel<<<BC_N, 128, 0, stream>>>(x, ws, out);
}